// HyperMLPHead_764504179422
// MI455X (gfx1250) — compile-verified
//
#include <hip/hip_runtime.h>
#include <math.h>

// Problem constants (match reference)
#define BB 2
#define TT 2048
#define DD 1024
#define RR 64
#define RSS 16
#define BT (BB*TT)          // 4096 rows when batch is flattened

typedef __attribute__((ext_vector_type(2))) float v2f;
typedef __attribute__((ext_vector_type(8))) float v8f;

// ---------------------------------------------------------------------------
// Generic batched fp32 WMMA GEMM:  C[b] = (A[b] @ B[b]') (* mulC[b])
// Block: 256(M) x 64(N), K-chunks of 32, double-buffered LDS.
// 256 threads = 8 wave32; wave w owns rows [w*32, w*32+32) x 64 cols
//   -> 2 M-tiles x 4 N-tiles = 8 accumulators of 16x16 f32.
// LDS layout: A row-major [256][34] (even pad -> each (k,k+1) fragment is one
// aligned ds_load_b64, stride-34 conflict-free); B stored transposed
// BsT[n][k] with the same property.
// Requires M%256==0, N%64==0, K%32==0 (true for every launch here).
// bT==1 means B is stored [N,K] (ldb strides over K) -> computes A @ B^T.
// ---------------------------------------------------------------------------
#define GBM 256
#define GBN 64
#define GBK 32

__global__ __launch_bounds__(256) void gemm_wmma_f32(
    const float* __restrict__ A, const float* __restrict__ Bm,
    float* __restrict__ C, const float* __restrict__ mulC,
    int M, int N, int K, int lda, int ldb, int ldc, int bT,
    long long sA, long long sB, long long sC)
{
    __shared__ __align__(16) float As [2][GBM][34];
    __shared__ __align__(16) float BsT[2][GBN][34];

    const int tid  = threadIdx.x;
    const int wave = tid >> 5;
    const int lane = tid & 31;
    const int hi   = lane >> 4;     // 0: K pair {k,k+1}; 1: {k+2,k+3}
    const int lm   = lane & 15;

    const int m0 = blockIdx.y * GBM;
    const int n0 = blockIdx.x * GBN;

    const float* Ab = A  + (size_t)blockIdx.z * sA;
    const float* Bb = Bm + (size_t)blockIdx.z * sB;
    float*       Cb = C  + (size_t)blockIdx.z * sC;
    const float* Mb = mulC ? (mulC + (size_t)blockIdx.z * sC) : nullptr;

    // per-thread staging registers (register prefetch for double buffering)
    float4 ra[8];   // A tile: 256x32 = 2048 float4 / 256 thr = 8 each
    float4 rb[2];   // B tile:  32x64 =  512 float4 / 256 thr = 2 each

    // Precomputed load coordinates
    int arow[8], ac4[8];
    #pragma unroll
    for (int i = 0; i < 8; ++i) { int idx = tid + i * 256; arow[i] = idx >> 3; ac4[i] = (idx & 7) * 4; }
    int bi0[2], bi1[2];
    #pragma unroll
    for (int i = 0; i < 2; ++i) {
        int idx = tid + i * 256;
        if (true) { /* both mappings derived from idx below */ }
        bi0[i] = idx; bi1[i] = idx;
    }

    auto loadGlobal = [&](int k0) {
        #pragma unroll
        for (int i = 0; i < 8; ++i)
            ra[i] = *(const float4*)(Ab + (size_t)(m0 + arow[i]) * lda + k0 + ac4[i]);
        if (bT) {
            #pragma unroll
            for (int i = 0; i < 2; ++i) {
                int idx = bi0[i];
                int n = idx >> 3, c4 = (idx & 7) * 4;          // coalesced along K
                rb[i] = *(const float4*)(Bb + (size_t)(n0 + n) * ldb + k0 + c4);
            }
        } else {
            #pragma unroll
            for (int i = 0; i < 2; ++i) {
                int idx = bi1[i];
                int r = idx >> 4, c4 = (idx & 15) * 4;         // coalesced along N
                rb[i] = *(const float4*)(Bb + (size_t)(k0 + r) * ldb + n0 + c4);
            }
        }
    };

    auto storeLDS = [&](int buf) {
        #pragma unroll
        for (int i = 0; i < 8; ++i) {
            float4 t = ra[i];
            v2f lo; lo.x = t.x; lo.y = t.y;
            v2f hi2; hi2.x = t.z; hi2.y = t.w;
            *(v2f*)&As[buf][arow[i]][ac4[i]]     = lo;
            *(v2f*)&As[buf][arow[i]][ac4[i] + 2] = hi2;
        }
        if (bT) {
            #pragma unroll
            for (int i = 0; i < 2; ++i) {
                int idx = bi0[i];
                int n = idx >> 3, c4 = (idx & 7) * 4;
                float4 t = rb[i];
                v2f lo; lo.x = t.x; lo.y = t.y;
                v2f hi2; hi2.x = t.z; hi2.y = t.w;
                *(v2f*)&BsT[buf][n][c4]     = lo;
                *(v2f*)&BsT[buf][n][c4 + 2] = hi2;
            }
        } else {
            #pragma unroll
            for (int i = 0; i < 2; ++i) {
                int idx = bi1[i];
                int r = idx >> 4, c4 = (idx & 15) * 4;
                float4 t = rb[i];
                BsT[buf][c4 + 0][r] = t.x;
                BsT[buf][c4 + 1][r] = t.y;
                BsT[buf][c4 + 2][r] = t.z;
                BsT[buf][c4 + 3][r] = t.w;
            }
        }
    };

    v8f acc00 = {}, acc01 = {}, acc02 = {}, acc03 = {};
    v8f acc10 = {}, acc11 = {}, acc12 = {}, acc13 = {};

    const int r0 = wave * 32 + lm;         // first 16-row tile of this wave
    const int r1 = r0 + 16;                // second 16-row tile
    const int nchunks = K / GBK;

    loadGlobal(0);
    for (int c = 0; c < nchunks; ++c) {
        const int buf = c & 1;
        storeLDS(buf);
        __syncthreads();
        if (c + 1 < nchunks) loadGlobal((c + 1) * GBK);   // overlap with WMMA burst

        #pragma unroll
        for (int kk = 0; kk < GBK; kk += 4) {
            const int kc = kk + 2 * hi;
            v2f a0 = *(const v2f*)&As [buf][r0][kc];
            v2f a1 = *(const v2f*)&As [buf][r1][kc];
            v2f b0 = *(const v2f*)&BsT[buf][lm +  0][kc];
            v2f b1 = *(const v2f*)&BsT[buf][lm + 16][kc];
            v2f b2 = *(const v2f*)&BsT[buf][lm + 32][kc];
            v2f b3 = *(const v2f*)&BsT[buf][lm + 48][kc];
            acc00 = __builtin_amdgcn_wmma_f32_16x16x4_f32(false, a0, false, b0, (short)0, acc00, false, false);
            acc01 = __builtin_amdgcn_wmma_f32_16x16x4_f32(false, a0, false, b1, (short)0, acc01, false, false);
            acc02 = __builtin_amdgcn_wmma_f32_16x16x4_f32(false, a0, false, b2, (short)0, acc02, false, false);
            acc03 = __builtin_amdgcn_wmma_f32_16x16x4_f32(false, a0, false, b3, (short)0, acc03, false, false);
            acc10 = __builtin_amdgcn_wmma_f32_16x16x4_f32(false, a1, false, b0, (short)0, acc10, false, false);
            acc11 = __builtin_amdgcn_wmma_f32_16x16x4_f32(false, a1, false, b1, (short)0, acc11, false, false);
            acc12 = __builtin_amdgcn_wmma_f32_16x16x4_f32(false, a1, false, b2, (short)0, acc12, false, false);
            acc13 = __builtin_amdgcn_wmma_f32_16x16x4_f32(false, a1, false, b3, (short)0, acc13, false, false);
        }
        __syncthreads();
    }

    // C/D layout: VGPR r, lanes 0-15 -> M=r, lanes 16-31 -> M=r+8, N=lane%16
    const int ncol = n0 + lm;
    #pragma unroll
    for (int mt = 0; mt < 2; ++mt) {
        const int mrow = m0 + wave * 32 + mt * 16 + hi * 8;
        v8f a0 = mt ? acc10 : acc00, a1 = mt ? acc11 : acc01;
        v8f a2 = mt ? acc12 : acc02, a3 = mt ? acc13 : acc03;
        #pragma unroll
        for (int r = 0; r < 8; ++r) {
            size_t base = (size_t)(mrow + r) * ldc + ncol;
            float v0 = a0[r], v1 = a1[r], v2 = a2[r], v3 = a3[r];
            if (Mb) {
                v0 *= Mb[base]; v1 *= Mb[base + 16]; v2 *= Mb[base + 32]; v3 *= Mb[base + 48];
            }
            Cb[base] = v0; Cb[base + 16] = v1; Cb[base + 32] = v2; Cb[base + 48] = v3;
        }
    }
}

// ---------------------------------------------------------------------------
// Helpers
// ---------------------------------------------------------------------------
__device__ inline float waveReduce(float v) {
    #pragma unroll
    for (int o = 16; o > 0; o >>= 1) v += __shfl_xor(v, o, 32);
    return v;
}
__device__ inline float dot4f(float4 a, float4 b) {
    return a.x * b.x + a.y * b.y + a.z * b.z + a.w * b.w;
}
__device__ inline float softplusf(float x) {
    if (x > 20.f) return x;
    if (x < -20.f) return expf(x);
    return log1pf(expf(x));
}

// d = x@Wd [BT], A = x@Wa [BT,16], Bm = x@Wb [BT,16]. One wave per row.
__global__ __launch_bounds__(256) void proj_dab_kernel(
    const float* __restrict__ x, const float* __restrict__ Wd,
    const float* __restrict__ Wa, const float* __restrict__ Wb,
    float* __restrict__ dOut, float* __restrict__ aOut, float* __restrict__ bOut)
{
    const int wave = threadIdx.x >> 5, lane = threadIdx.x & 31;
    const int row  = blockIdx.x * 8 + wave;
    const float* xr = x + (size_t)row * DD;

    float xv[DD / 32];
    #pragma unroll
    for (int i = 0; i < DD / 32; ++i) xv[i] = xr[lane + 32 * i];

    float acc = 0.f;
    #pragma unroll
    for (int i = 0; i < DD / 32; ++i) acc += xv[i] * Wd[lane + 32 * i];
    acc = waveReduce(acc);
    if (lane == 0) dOut[row] = acc;

    for (int c = 0; c < RSS; ++c) {
        float a = 0.f, bm = 0.f;
        #pragma unroll
        for (int i = 0; i < DD / 32; ++i) {
            int k = lane + 32 * i;
            a  += xv[i] * Wa[(size_t)k * RSS + c];
            bm += xv[i] * Wb[(size_t)k * RSS + c];
        }
        a = waveReduce(a); bm = waveReduce(bm);
        if (lane == 0) { aOut[(size_t)row * RSS + c] = a; bOut[(size_t)row * RSS + c] = bm; }
    }
}

// zB[b,i,:] = sum_j z[b,i,j] * Bm[b,j,:].  One wave per (b,i) row.
__global__ __launch_bounds__(256) void zmatB_kernel(
    const float* __restrict__ z, const float* __restrict__ Bm, float* __restrict__ zB)
{
    const int wave = threadIdx.x >> 5, lane = threadIdx.x & 31;
    const int rowg = blockIdx.x * 8 + wave;          // b*T + i
    const int b    = rowg >> 11;                     // T = 2048
    const float* zr = z + (size_t)rowg * TT;
    const float* Bb = Bm + (size_t)b * TT * RSS;

    float acc[RSS];
    #pragma unroll
    for (int r = 0; r < RSS; ++r) acc[r] = 0.f;

    for (int j = lane; j < TT; j += 32) {
        float zv = zr[j];
        const float4* br = (const float4*)(Bb + (size_t)j * RSS);
        float4 b0 = br[0], b1 = br[1], b2 = br[2], b3 = br[3];
        acc[0]  += zv * b0.x; acc[1]  += zv * b0.y; acc[2]  += zv * b0.z; acc[3]  += zv * b0.w;
        acc[4]  += zv * b1.x; acc[5]  += zv * b1.y; acc[6]  += zv * b1.z; acc[7]  += zv * b1.w;
        acc[8]  += zv * b2.x; acc[9]  += zv * b2.y; acc[10] += zv * b2.z; acc[11] += zv * b2.w;
        acc[12] += zv * b3.x; acc[13] += zv * b3.y; acc[14] += zv * b3.z; acc[15] += zv * b3.w;
    }
    #pragma unroll
    for (int r = 0; r < RSS; ++r) {
        float v = waveReduce(acc[r]);
        if (lane == 0) zB[(size_t)rowg * RSS + r] = v;
    }
}

// out[b,i,j] = z[b,i,j]*d[b,j] + dot16(zB[b,i,:], A[b,j,:]). In-place safe.
__global__ __launch_bounds__(256) void dplr_combine_kernel(
    const float* __restrict__ z, const float* __restrict__ dvec,
    const float* __restrict__ zB, const float* __restrict__ Amat,
    float* __restrict__ out)
{
    const int rowg = blockIdx.y;                 // b*T + i
    const int b    = rowg >> 11;
    const int j    = (blockIdx.x * blockDim.x + threadIdx.x) * 4;

    const float4* cr = (const float4*)(zB + (size_t)rowg * RSS);
    const float4 c0 = cr[0], c1 = cr[1], c2 = cr[2], c3 = cr[3];

    const float4 z4 = *(const float4*)(z + (size_t)rowg * TT + j);
    const float4 d4 = *(const float4*)(dvec + (size_t)b * TT + j);

    const float4* A0 = (const float4*)(Amat + ((size_t)b * TT + j) * RSS);
    float dv[4];
    #pragma unroll
    for (int q = 0; q < 4; ++q) {
        const float4* Ar = A0 + q * 4;
        dv[q] = dot4f(c0, Ar[0]) + dot4f(c1, Ar[1]) + dot4f(c2, Ar[2]) + dot4f(c3, Ar[3]);
    }
    float4 o;
    o.x = z4.x * d4.x + dv[0];
    o.y = z4.y * d4.y + dv[1];
    o.z = z4.z * d4.z + dv[2];
    o.w = z4.w * d4.w + dv[3];
    *(float4*)(out + (size_t)rowg * TT + j) = o;
}

// denom[row] = ||h[row,:]||_2 + 1e-8. One wave per row.
__global__ __launch_bounds__(256) void rownorm_kernel(
    const float* __restrict__ h, float* __restrict__ denom)
{
    const int wave = threadIdx.x >> 5, lane = threadIdx.x & 31;
    const int rowg = blockIdx.x * 8 + wave;
    const float* hr = h + (size_t)rowg * TT;
    float acc = 0.f;
    for (int j = lane; j < TT; j += 32) { float v = hr[j]; acc += v * v; }
    acc = waveReduce(acc);
    if (lane == 0) denom[rowg] = sqrtf(acc) + 1e-8f;
}

// h = softplus(h_scale) * relu(h_mix / denom[row]); writes over h_mix.
__global__ __launch_bounds__(256) void combine_h_kernel(
    float* __restrict__ hmix, const float* __restrict__ hscale,
    const float* __restrict__ denom)
{
    const int rowg = blockIdx.y;
    const int j = (blockIdx.x * blockDim.x + threadIdx.x) * 4;
    const float inv = 1.f / denom[rowg];
    float4 m = *(const float4*)(hmix   + (size_t)rowg * TT + j);
    float4 s = *(const float4*)(hscale + (size_t)rowg * TT + j);
    float4 o;
    o.x = softplusf(s.x) * fmaxf(m.x * inv, 0.f);
    o.y = softplusf(s.y) * fmaxf(m.y * inv, 0.f);
    o.z = softplusf(s.z) * fmaxf(m.z * inv, 0.f);
    o.w = softplusf(s.w) * fmaxf(m.w * inv, 0.f);
    *(float4*)(hmix + (size_t)rowg * TT + j) = o;
}

// a *= b elementwise (float4)
__global__ __launch_bounds__(256) void emul_kernel(float* __restrict__ a, const float* __restrict__ b)
{
    const size_t i = ((size_t)blockIdx.x * blockDim.x + threadIdx.x) * 4;
    float4 av = *(const float4*)(a + i);
    float4 bv = *(const float4*)(b + i);
    av.x *= bv.x; av.y *= bv.y; av.z *= bv.z; av.w *= bv.w;
    *(float4*)(a + i) = av;
}

// ---------------------------------------------------------------------------
extern "C" void kernel_launch(void* const* d_in, const int* in_sizes, int n_in,
                              void* d_out, int out_size, void* d_ws, size_t ws_size,
                              hipStream_t stream) {
    (void)in_sizes; (void)n_in; (void)out_size; (void)ws_size;

    const float* x = (const float*)d_in[0];
    const float* qk_Wq1  = (const float*)d_in[1];  const float* qk_Wq2  = (const float*)d_in[2];
    const float* qk_Wm1  = (const float*)d_in[3];  const float* qk_Wm2  = (const float*)d_in[4];
    const float* qk_Wk1  = (const float*)d_in[5];  const float* qk_Wk2  = (const float*)d_in[6];
    const float* qks_Wq1 = (const float*)d_in[7];  const float* qks_Wq2 = (const float*)d_in[8];
    const float* qks_Wm1 = (const float*)d_in[9];  const float* qks_Wm2 = (const float*)d_in[10];
    const float* qks_Wk1 = (const float*)d_in[11]; const float* qks_Wk2 = (const float*)d_in[12];
    const float* vo_Wv1  = (const float*)d_in[13]; const float* vo_Wv2  = (const float*)d_in[14];
    const float* vo_Wm1  = (const float*)d_in[15]; const float* vo_Wm2  = (const float*)d_in[16];
    const float* vo_Wo   = (const float*)d_in[17];
    const float* sm1_Wd  = (const float*)d_in[18]; const float* sm1_Wa  = (const float*)d_in[19]; const float* sm1_Wb  = (const float*)d_in[20];
    const float* sm1s_Wd = (const float*)d_in[21]; const float* sm1s_Wa = (const float*)d_in[22]; const float* sm1s_Wb = (const float*)d_in[23];
    const float* sm2_Wd  = (const float*)d_in[24]; const float* sm2_Wa  = (const float*)d_in[25]; const float* sm2_Wb  = (const float*)d_in[26];

    const size_t BTD = (size_t)BT * DD;     // 4,194,304
    const size_t BTT = (size_t)BB * TT * TT;

    float* w = (float*)d_ws;
    float* qg   = w; w += BTD;
    float* kb   = w; w += BTD;
    float* qsg  = w; w += BTD;
    float* ksb  = w; w += BTD;
    float* vb   = w; w += BTD;
    float* mvob = w; w += BTD;
    float* tmpb = w; w += BTD;              // mg / mgs scratch
    float* zb   = w; w += BTT;              // z -> h_mix -> h
    float* zsb  = w; w += BTT;              // zs -> h_scale -> readout
    float* ctx  = w; w += BTD;
    float* xr   = w; w += (size_t)BT * RR;
    float* d1   = w; w += BT; float* d1s = w; w += BT; float* d2 = w; w += BT;
    float* A1   = w; w += (size_t)BT * RSS; float* B1  = w; w += (size_t)BT * RSS;
    float* A1s  = w; w += (size_t)BT * RSS; float* B1s = w; w += (size_t)BT * RSS;
    float* A2   = w; w += (size_t)BT * RSS; float* B2  = w; w += (size_t)BT * RSS;
    float* zBb  = w; w += (size_t)BT * RSS;
    float* denom= w; w += BT;

    auto gemm = [&](const float* A, const float* Bm, float* C, const float* mul,
                    int M, int N, int K, int lda, int ldb, int ldc, int bT,
                    long long sA, long long sB, long long sC, int batches) {
        dim3 g((unsigned)(N / GBN), (unsigned)(M / GBM), (unsigned)batches);
        gemm_wmma_f32<<<g, dim3(256), 0, stream>>>(A, Bm, C, mul, M, N, K, lda, ldb, ldc, bT, sA, sB, sC);
    };

    // ---- DPLR d/A/B projections (3 paths) ----
    proj_dab_kernel<<<dim3(BT / 8), 256, 0, stream>>>(x, sm1_Wd,  sm1_Wa,  sm1_Wb,  d1,  A1,  B1);
    proj_dab_kernel<<<dim3(BT / 8), 256, 0, stream>>>(x, sm1s_Wd, sm1s_Wa, sm1s_Wb, d1s, A1s, B1s);
    proj_dab_kernel<<<dim3(BT / 8), 256, 0, stream>>>(x, sm2_Wd,  sm2_Wa,  sm2_Wb,  d2,  A2,  B2);

    // ---- Low-rank feature mixing (8 projections; gates fused via epilogue) ----
    gemm(x,  qk_Wm1, xr,   nullptr, BT, RR, DD, DD, RR, RR, 0, 0, 0, 0, 1);
    gemm(xr, qk_Wm2, tmpb, nullptr, BT, DD, RR, RR, DD, DD, 0, 0, 0, 0, 1);
    gemm(x,  qk_Wq1, xr,   nullptr, BT, RR, DD, DD, RR, RR, 0, 0, 0, 0, 1);
    gemm(xr, qk_Wq2, qg,   tmpb,    BT, DD, RR, RR, DD, DD, 0, 0, 0, 0, 1);
    gemm(x,  qk_Wk1, xr,   nullptr, BT, RR, DD, DD, RR, RR, 0, 0, 0, 0, 1);
    gemm(xr, qk_Wk2, kb,   nullptr, BT, DD, RR, RR, DD, DD, 0, 0, 0, 0, 1);

    gemm(x,  qks_Wm1, xr,   nullptr, BT, RR, DD, DD, RR, RR, 0, 0, 0, 0, 1);
    gemm(xr, qks_Wm2, tmpb, nullptr, BT, DD, RR, RR, DD, DD, 0, 0, 0, 0, 1);
    gemm(x,  qks_Wq1, xr,   nullptr, BT, RR, DD, DD, RR, RR, 0, 0, 0, 0, 1);
    gemm(xr, qks_Wq2, qsg,  tmpb,    BT, DD, RR, RR, DD, DD, 0, 0, 0, 0, 1);
    gemm(x,  qks_Wk1, xr,   nullptr, BT, RR, DD, DD, RR, RR, 0, 0, 0, 0, 1);
    gemm(xr, qks_Wk2, ksb,  nullptr, BT, DD, RR, RR, DD, DD, 0, 0, 0, 0, 1);

    gemm(x,  vo_Wv1, xr,   nullptr, BT, RR, DD, DD, RR, RR, 0, 0, 0, 0, 1);
    gemm(xr, vo_Wv2, vb,   nullptr, BT, DD, RR, RR, DD, DD, 0, 0, 0, 0, 1);
    gemm(x,  vo_Wm1, xr,   nullptr, BT, RR, DD, DD, RR, RR, 0, 0, 0, 0, 1);
    gemm(xr, vo_Wm2, mvob, nullptr, BT, DD, RR, RR, DD, DD, 0, 0, 0, 0, 1);

    // ---- Scores: z = qg @ k^T, zs = qsg @ ks^T  (batched over B) ----
    const long long sTD = (long long)TT * DD, sTTl = (long long)TT * TT;
    gemm(qg,  kb,  zb,  nullptr, TT, TT, DD, DD, DD, TT, 1, sTD, sTD, sTTl, BB);
    gemm(qsg, ksb, zsb, nullptr, TT, TT, DD, DD, DD, TT, 1, sTD, sTD, sTTl, BB);

    // ---- DPLR mixing 1 (gate + scale), norm, HyperGLU combine ----
    zmatB_kernel<<<dim3(BT / 8), 256, 0, stream>>>(zb, B1, zBb);
    dplr_combine_kernel<<<dim3(TT / 1024, BT), 256, 0, stream>>>(zb, d1, zBb, A1, zb);
    zmatB_kernel<<<dim3(BT / 8), 256, 0, stream>>>(zsb, B1s, zBb);
    dplr_combine_kernel<<<dim3(TT / 1024, BT), 256, 0, stream>>>(zsb, d1s, zBb, A1s, zsb);
    rownorm_kernel<<<dim3(BT / 8), 256, 0, stream>>>(zb, denom);
    combine_h_kernel<<<dim3(TT / 1024, BT), 256, 0, stream>>>(zb, zsb, denom);

    // ---- DPLR mixing 2 -> readout (into zsb) ----
    zmatB_kernel<<<dim3(BT / 8), 256, 0, stream>>>(zb, B2, zBb);
    dplr_combine_kernel<<<dim3(TT / 1024, BT), 256, 0, stream>>>(zb, d2, zBb, A2, zsb);

    // ---- context = readout @ v; out = (context * mvo) @ Wo^T ----
    gemm(zsb, vb, ctx, nullptr, TT, DD, TT, TT, DD, DD, 0, sTTl, sTD, sTD, BB);
    emul_kernel<<<dim3((unsigned)(BTD / 4 / 256)), 256, 0, stream>>>(ctx, mvob);
    gemm(ctx, vo_Wo, (float*)d_out, nullptr, BT, DD, DD, DD, DD, DD, 1, 0, 0, 0, 1);
}